// BaselineGINModel_59871844106318
// MI455X (gfx1250) — compile-verified
//
#include <hip/hip_runtime.h>
#include <hip/hip_bf16.h>

typedef __attribute__((ext_vector_type(2))) float v2f;
typedef __attribute__((ext_vector_type(8))) float v8f;

#define HID 32

// ---------------------------------------------------------------------------
// out[M x 32] = act(in[M x K] @ w[K x 32] (+ bias)) using V_WMMA_F32_16X16X4_F32
// Requirements: M % 16 == 0, K % 4 == 0, K <= 128. One 16x16 D tile per wave.
// A fragment layout (16x4 f32): lanes 0-15 -> M=lane, K = {0,1}; lanes 16-31 ->
// M=lane-16, K = {2,3}. B (4x16): VGPR0 = rows K0/K2 striped over lane halves,
// VGPR1 = rows K1/K3. C/D (16x16 f32): VGPR r holds M=r (lanes 0-15) and M=r+8
// (lanes 16-31), N = lane&15.
// ---------------------------------------------------------------------------
__global__ __launch_bounds__(256)
void gin_gemm_wmma(const float* __restrict__ in, const float* __restrict__ w,
                   const float* __restrict__ bias, float* __restrict__ out,
                   int M, int K, int doRelu)
{
  __shared__ float wlds[128 * HID];           // up to 16 KB of weights in LDS
  const int tid = threadIdx.x;
  for (int i = tid; i < K * HID; i += 256) wlds[i] = w[i];
  __syncthreads();

  const int nTiles = (M >> 4) * 2;            // (M/16) x (32/16)
  const int gt = blockIdx.x * 8 + (tid >> 5); // 8 waves / block
  if (gt >= nTiles) return;                    // wave-uniform: EXEC stays all-1s

  const int mt   = gt >> 1;
  const int nt   = gt & 1;
  const int lane = tid & 31;
  const int l15  = lane & 15;
  const int hi   = lane >> 4;                  // 0 or 1 (lane half)
  const int kq   = hi * 2;                     // K sub-offset inside 4-wide step
  const int row  = mt * 16 + l15;              // A row this lane supplies
  const int col  = nt * 16 + l15;              // B/D column this lane supplies

  v8f c = {};
  const float* arow = in + (long long)row * K;
  for (int k = 0; k < K; k += 4) {
    const float2 av = *(const float2*)(arow + k + kq);   // 8B-aligned pair
    v2f a; a.x = av.x; a.y = av.y;
    v2f b; b.x = wlds[(k + kq) * HID + col];
           b.y = wlds[(k + kq + 1) * HID + col];
    // D = A(16x4) * B(4x16) + C  -- full f32 precision
    c = __builtin_amdgcn_wmma_f32_16x16x4_f32(false, a, false, b,
                                              (short)0, c, false, false);
  }

  const float bv = bias ? bias[col] : 0.0f;
  const int rbase = mt * 16 + hi * 8;
  #pragma unroll
  for (int r = 0; r < 8; ++r) {
    float v = c[r] + bv;
    if (doRelu) v = fmaxf(v, 0.0f);
    out[(long long)(rbase + r) * HID + col] = v;
  }
}

// ---------------------------------------------------------------------------
// agg[dst[e], :] += t[src[e], :]   (8 threads per edge, float4 chunks)
// ---------------------------------------------------------------------------
__global__ __launch_bounds__(256)
void gin_scatter_add(const float* __restrict__ t, const int* __restrict__ ei,
                     float* __restrict__ agg, int nEdges)
{
  const int tid = blockIdx.x * 256 + threadIdx.x;
  if (tid >= nEdges * 8) return;
  const int e = tid >> 3;
  const int c = (tid & 7) * 4;
  const int s = ei[e];               // edge_index[0][e]
  const int d = ei[nEdges + e];      // edge_index[1][e]
  const float4 v = *(const float4*)(t + (long long)s * HID + c);
  float* p = agg + (long long)d * HID + c;
  atomicAdd(p + 0, v.x);
  atomicAdd(p + 1, v.y);
  atomicAdd(p + 2, v.z);
  atomicAdd(p + 3, v.w);
}

// ---------------------------------------------------------------------------
// agg = relu(agg + t + bias[col])  (in place, float4)
// ---------------------------------------------------------------------------
__global__ __launch_bounds__(256)
void gin_combine_relu(float* __restrict__ agg, const float* __restrict__ t,
                      const float* __restrict__ bias, int n4)
{
  const int i = blockIdx.x * 256 + threadIdx.x;
  if (i >= n4) return;
  const int cb = (i * 4) & (HID - 1);          // column of first element
  const float4 a = ((const float4*)agg)[i];
  const float4 b = ((const float4*)t)[i];
  float4 r;
  r.x = fmaxf(a.x + b.x + bias[cb + 0], 0.0f);
  r.y = fmaxf(a.y + b.y + bias[cb + 1], 0.0f);
  r.z = fmaxf(a.z + b.z + bias[cb + 2], 0.0f);
  r.w = fmaxf(a.w + b.w + bias[cb + 3], 0.0f);
  ((float4*)agg)[i] = r;
}

// ---------------------------------------------------------------------------
// sums[batch[n], :] += h[n, :]
// ---------------------------------------------------------------------------
__global__ __launch_bounds__(256)
void gin_pool_sum(const float* __restrict__ h, const int* __restrict__ batch,
                  float* __restrict__ sums, int nNodes)
{
  const int tid = blockIdx.x * 256 + threadIdx.x;
  if (tid >= nNodes * 8) return;
  const int n = tid >> 3;
  const int c = (tid & 7) * 4;
  const int g = batch[n];
  const float4 v = *(const float4*)(h + (long long)n * HID + c);
  float* p = sums + (long long)g * HID + c;
  atomicAdd(p + 0, v.x);
  atomicAdd(p + 1, v.y);
  atomicAdd(p + 2, v.z);
  atomicAdd(p + 3, v.w);
}

__global__ __launch_bounds__(256)
void gin_pool_count(const int* __restrict__ batch, float* __restrict__ counts,
                    int nNodes)
{
  const int n = blockIdx.x * 256 + threadIdx.x;
  if (n < nNodes) atomicAdd(&counts[batch[n]], 1.0f);
}

// ---------------------------------------------------------------------------
// mean pool -> relu(p@wc1+bc1) -> @wc2+bc2 -> log_softmax   (512 graphs)
// ---------------------------------------------------------------------------
__global__ __launch_bounds__(256)
void gin_classifier(const float* __restrict__ sums, const float* __restrict__ counts,
                    const float* __restrict__ wc1, const float* __restrict__ bc1,
                    const float* __restrict__ wc2, const float* __restrict__ bc2,
                    float* __restrict__ out, int G)
{
  __shared__ float s1[HID * HID];
  __shared__ float s2[HID * 2];
  __shared__ float sb1[HID];
  __shared__ float sb2[2];
  const int tid = threadIdx.x;
  for (int i = tid; i < HID * HID; i += 256) s1[i] = wc1[i];
  for (int i = tid; i < HID * 2;   i += 256) s2[i] = wc2[i];
  if (tid < HID) sb1[tid] = bc1[tid];
  if (tid < 2)   sb2[tid] = bc2[tid];
  __syncthreads();

  for (int g = tid; g < G; g += 256) {
    const float inv = 1.0f / fmaxf(counts[g], 1.0f);
    float p[HID];
    #pragma unroll
    for (int j = 0; j < HID; ++j) p[j] = sums[g * HID + j] * inv;
    float u[HID];
    #pragma unroll
    for (int j = 0; j < HID; ++j) u[j] = sb1[j];
    for (int k = 0; k < HID; ++k) {
      const float pk = p[k];
      #pragma unroll
      for (int j = 0; j < HID; ++j) u[j] += pk * s1[k * HID + j];
    }
    float z0 = sb2[0], z1 = sb2[1];
    #pragma unroll
    for (int k = 0; k < HID; ++k) {
      const float uk = fmaxf(u[k], 0.0f);
      z0 += uk * s2[k * 2 + 0];
      z1 += uk * s2[k * 2 + 1];
    }
    const float m = fmaxf(z0, z1);
    const float l = m + logf(expf(z0 - m) + expf(z1 - m));
    out[g * 2 + 0] = z0 - l;
    out[g * 2 + 1] = z1 - l;
  }
}

// ---------------------------------------------------------------------------
extern "C" void kernel_launch(void* const* d_in, const int* in_sizes, int n_in,
                              void* d_out, int out_size, void* d_ws, size_t ws_size,
                              hipStream_t stream)
{
  const float* x   = (const float*)d_in[0];
  const float* w1a = (const float*)d_in[1];
  const float* b1a = (const float*)d_in[2];
  const float* w1b = (const float*)d_in[3];
  const float* b1b = (const float*)d_in[4];
  const float* w2a = (const float*)d_in[5];
  const float* b2a = (const float*)d_in[6];
  const float* w2b = (const float*)d_in[7];
  const float* b2b = (const float*)d_in[8];
  const float* w3a = (const float*)d_in[9];
  const float* b3a = (const float*)d_in[10];
  const float* w3b = (const float*)d_in[11];
  const float* b3b = (const float*)d_in[12];
  const float* wc1 = (const float*)d_in[13];
  const float* bc1 = (const float*)d_in[14];
  const float* wc2 = (const float*)d_in[15];
  const float* bc2 = (const float*)d_in[16];
  const int* ei    = (const int*)d_in[17];
  const int* batch = (const int*)d_in[18];

  const int M  = in_sizes[18];          // 100000 nodes (divisible by 16)
  const int E  = in_sizes[17] / 2;      // 1.6M edges
  const int IN = in_sizes[0] / M;       // 128
  const int G  = out_size / 2;          // 512 graphs
  float* outF  = (float*)d_out;

  // workspace: 3 node buffers + pooled sums + counts  (~38.5 MB)
  float* A      = (float*)d_ws;                  // t  = h @ Wa
  float* B      = A + (size_t)M * HID;           // agg -> u
  float* C      = B + (size_t)M * HID;           // h
  float* sums   = C + (size_t)M * HID;           // G x 32
  float* counts = sums + (size_t)G * HID;        // G

  const int nTiles   = (M / 16) * 2;
  const int gBlocks  = (nTiles + 7) / 8;
  const int eBlocks  = (E * 8 + 255) / 256;
  const int c4Blocks = (M * HID / 4 + 255) / 256;
  const int pBlocks  = (M * 8 + 255) / 256;
  const size_t nodeBytes = (size_t)M * HID * sizeof(float);

  // ---- layer 1 (128 -> 32, linear pushed below the segment_sum) ----
  gin_gemm_wmma<<<gBlocks, 256, 0, stream>>>(x, w1a, nullptr, A, M, IN, 0);
  hipMemsetAsync(B, 0, nodeBytes, stream);
  gin_scatter_add<<<eBlocks, 256, 0, stream>>>(A, ei, B, E);
  gin_combine_relu<<<c4Blocks, 256, 0, stream>>>(B, A, b1a, M * HID / 4);
  gin_gemm_wmma<<<gBlocks, 256, 0, stream>>>(B, w1b, b1b, C, M, HID, 1);

  // ---- layer 2 ----
  gin_gemm_wmma<<<gBlocks, 256, 0, stream>>>(C, w2a, nullptr, A, M, HID, 0);
  hipMemsetAsync(B, 0, nodeBytes, stream);
  gin_scatter_add<<<eBlocks, 256, 0, stream>>>(A, ei, B, E);
  gin_combine_relu<<<c4Blocks, 256, 0, stream>>>(B, A, b2a, M * HID / 4);
  gin_gemm_wmma<<<gBlocks, 256, 0, stream>>>(B, w2b, b2b, C, M, HID, 1);

  // ---- layer 3 ----
  gin_gemm_wmma<<<gBlocks, 256, 0, stream>>>(C, w3a, nullptr, A, M, HID, 0);
  hipMemsetAsync(B, 0, nodeBytes, stream);
  gin_scatter_add<<<eBlocks, 256, 0, stream>>>(A, ei, B, E);
  gin_combine_relu<<<c4Blocks, 256, 0, stream>>>(B, A, b3a, M * HID / 4);
  gin_gemm_wmma<<<gBlocks, 256, 0, stream>>>(B, w3b, b3b, C, M, HID, 1);

  // ---- global mean pool + classifier + log_softmax ----
  hipMemsetAsync(sums, 0, (size_t)(G * HID + G) * sizeof(float), stream);
  gin_pool_sum<<<pBlocks, 256, 0, stream>>>(C, batch, sums, M);
  gin_pool_count<<<(M + 255) / 256, 256, 0, stream>>>(batch, counts, M);
  gin_classifier<<<1, 256, 0, stream>>>(sums, counts, wc1, bc1, wc2, bc2, outF, G);
}